// JointIntegralRegressor_9594956939914
// MI455X (gfx1250) — compile-verified
//
#include <hip/hip_runtime.h>
#include <math.h>

// Problem constants (reference: heatmaps [16, 24, 64, 64, 64] fp32)
#define NMAPS 384                  // 16 * 24 heatmaps
#define VOL   (64 * 64 * 64)       // 262144 elements per heatmap
#define SPLIT 8                    // chunks per heatmap -> 3072 blocks
#define CHUNK (VOL / SPLIT)        // 32768 elements per block (128 KB)
#define TPB   256                  // 8 wave32s per block
#define TILE  4096                 // elements per TDM tile (16 KB)
#define NTILES (CHUNK / TILE)      // 8 tiles per block
#define VPT_TILE (TILE / (TPB * 4))// 4 float4 per thread per tile

typedef __attribute__((ext_vector_type(4))) float        vf4;
typedef __attribute__((ext_vector_type(4))) unsigned int u32x4;
typedef __attribute__((ext_vector_type(8))) int          i32x8;
typedef __attribute__((ext_vector_type(4))) int          i32x4;

// Toolchain arity switch: the TDM header ships only with the 6-arg toolchain.
#if __has_include(<hip/amd_detail/amd_gfx1250_TDM.h>)
#define TDM_ARGS6 1
#else
#define TDM_ARGS6 0
#endif

// Post one asynchronous 1-D TDM transfer: TILE fp32 elements, global -> LDS.
// D# group0: count=1 | lds_addr | global_addr[56:0] | type=2 ("image").
// D# group1: wg_mask=0, data_size=2 (4B), tensor_dim0=tile_dim0=TILE,
//            tensor_dim1=tile_dim1=1, tensor_dim0_stride=TILE. Groups 2/3 = 0.
__device__ __forceinline__ void tdm_load_tile(unsigned lds_byte,
                                              unsigned long long gaddr) {
    u32x4 g0;
    g0.x = 1u;                                           // count=1 (valid user D#)
    g0.y = lds_byte;                                     // lds_addr (bytes)
    g0.z = (unsigned)(gaddr & 0xFFFFFFFFull);            // global_addr[31:0]
    g0.w = (unsigned)((gaddr >> 32) & 0x01FFFFFFull)     // global_addr[56:32]
         | 0x80000000u;                                  // type = 2 -> bits[127:126]
    i32x8 g1 = {0, 0, 0, 0, 0, 0, 0, 0};
    g1[0] = 0x00020000;                     // data_size = 2 (4 bytes), wg_mask = 0
    g1[1] = (int)(((unsigned)TILE) << 16);  // tensor_dim0[15:0] @ bits[63:48]
    g1[2] = 0x00010000;                     // tensor_dim0[31:16]=0 | tensor_dim1=1
    g1[3] = (int)(((unsigned)TILE) << 16);  // tile_dim0 @ bits[127:112]
    g1[4] = 1;                              // tile_dim1 = 1
    g1[5] = TILE;                           // tensor_dim0_stride[31:0]
    i32x4 gz = {0, 0, 0, 0};                // groups 2/3: higher dims unused
#if TDM_ARGS6
    i32x8 gz8 = {0, 0, 0, 0, 0, 0, 0, 0};
    __builtin_amdgcn_tensor_load_to_lds(g0, g1, gz, gz, gz8, 0);
#else
    __builtin_amdgcn_tensor_load_to_lds(g0, g1, gz, gz, 0);
#endif
}

// Online-softmax combine of two partial tuples (m,s,sx,sy,sz).
__device__ __forceinline__ void osm_combine(float& m, float& s, float& sx, float& sy, float& sz,
                                            float om, float os, float osx, float osy, float osz) {
    float nm = fmaxf(m, om);
    float a = __expf(m - nm);
    float b = __expf(om - nm);
    s  = s  * a + os  * b;
    sx = sx * a + osx * b;
    sy = sy * a + osy * b;
    sz = sz * a + osz * b;
    m = nm;
}

__global__ void __launch_bounds__(TPB)
jir_partial_kernel(const float* __restrict__ hm, float* __restrict__ ws) {
    __shared__ float buf[2][TILE];           // 32 KB double buffer in LDS

    const int blk   = blockIdx.x;            // [0, NMAPS*SPLIT)
    const int map   = blk / SPLIT;
    const int chunk = blk % SPLIT;
    const int tid   = threadIdx.x;

    const float* gsrc = hm + (size_t)map * VOL + (size_t)chunk * CHUNK;
    const unsigned long long gbase = (unsigned long long)(uintptr_t)gsrc;
    const unsigned lds0 = (unsigned)(uintptr_t)(void*)&buf[0][0];
    const unsigned lds1 = (unsigned)(uintptr_t)(void*)&buf[1][0];

    // Prologue: wave 0 posts tile 0 to the TDM engine.
    if (tid < 32) tdm_load_tile(lds0, gbase);

    float m = -INFINITY, s = 0.f, sx = 0.f, sy = 0.f, sz = 0.f;

    for (int t = 0; t < NTILES; ++t) {
        if (tid < 32) {                      // wave 0 runs the DMA pipeline
            if (t + 1 < NTILES) {
                tdm_load_tile(((t + 1) & 1) ? lds1 : lds0,
                              gbase + (unsigned long long)(t + 1) * (TILE * 4ull));
                __builtin_amdgcn_s_wait_tensorcnt(1);   // tile t complete (in-order)
            } else {
                __builtin_amdgcn_s_wait_tensorcnt(0);   // last tile complete
            }
        }
        __syncthreads();                     // publish tile t to all 8 waves

        const vf4* __restrict__ src = (const vf4*)buf[t & 1];
        #pragma unroll
        for (int i = 0; i < VPT_TILE; ++i) {
            const int vi = i * TPB + tid;             // float4 index within tile
            vf4 x = src[vi];                          // ds_load_b128
            const int v = chunk * CHUNK + t * TILE + vi * 4; // flat index in 64^3
            const float w0 = (float)(v & 63);         // W index of lane 0
            const float h  = (float)((v >> 6) & 63);  // H index (shared by 4)
            const float d  = (float)(v >> 12);        // D index (shared by 4)

            float nm = fmaxf(fmaxf(x.x, x.y), fmaxf(x.z, x.w));
            nm = fmaxf(m, nm);
            const float scale = __expf(m - nm);       // == 1 when max unchanged
            const float e0 = __expf(x.x - nm);
            const float e1 = __expf(x.y - nm);
            const float e2 = __expf(x.z - nm);
            const float e3 = __expf(x.w - nm);
            const float es = (e0 + e1) + (e2 + e3);

            s  = s  * scale + es;
            sx = sx * scale + (es * w0 + (e1 + 2.0f * e2 + 3.0f * e3));
            sy = sy * scale + es * h;
            sz = sz * scale + es * d;
            m  = nm;
        }
        __syncthreads();                     // all waves done with buf[t&1]
    }

    // Intra-wave reduction (wave32).
    #pragma unroll
    for (int k = 16; k >= 1; k >>= 1) {
        const float om  = __shfl_xor(m,  k, 32);
        const float os  = __shfl_xor(s,  k, 32);
        const float osx = __shfl_xor(sx, k, 32);
        const float osy = __shfl_xor(sy, k, 32);
        const float osz = __shfl_xor(sz, k, 32);
        osm_combine(m, s, sx, sy, sz, om, os, osx, osy, osz);
    }

    // Cross-wave reduction through LDS (8 waves per block).
    __shared__ float lm[8], ls[8], lsx[8], lsy[8], lsz[8];
    const int wave = tid >> 5, lane = tid & 31;
    if (lane == 0) { lm[wave] = m; ls[wave] = s; lsx[wave] = sx; lsy[wave] = sy; lsz[wave] = sz; }
    __syncthreads();

    if (tid < 8) {
        m = lm[tid]; s = ls[tid]; sx = lsx[tid]; sy = lsy[tid]; sz = lsz[tid];
        #pragma unroll
        for (int k = 4; k >= 1; k >>= 1) {
            const float om  = __shfl_xor(m,  k, 32);
            const float os  = __shfl_xor(s,  k, 32);
            const float osx = __shfl_xor(sx, k, 32);
            const float osy = __shfl_xor(sy, k, 32);
            const float osz = __shfl_xor(sz, k, 32);
            osm_combine(m, s, sx, sy, sz, om, os, osx, osy, osz);
        }
        if (tid == 0) {
            float* rec = ws + (size_t)blk * 5;
            rec[0] = m; rec[1] = s; rec[2] = sx; rec[3] = sy; rec[4] = sz;
        }
    }
}

__global__ void __launch_bounds__(64)
jir_final_kernel(const float* __restrict__ ws, float* __restrict__ out) {
    const int map = blockIdx.x * blockDim.x + threadIdx.x;
    if (map >= NMAPS) return;

    const float* rec = ws + (size_t)map * SPLIT * 5;
    float m = rec[0], s = rec[1], sx = rec[2], sy = rec[3], sz = rec[4];
    #pragma unroll
    for (int c = 1; c < SPLIT; ++c) {
        const float* r = rec + (size_t)c * 5;
        osm_combine(m, s, sx, sy, sz, r[0], r[1], r[2], r[3], r[4]);
    }

    // E[idx]/dim - 0.5 for each axis; all dims are 64.
    const float inv = 1.0f / (s * 64.0f);
    out[map * 3 + 0] = sx * inv - 0.5f;  // x (W axis)
    out[map * 3 + 1] = sy * inv - 0.5f;  // y (H axis)
    out[map * 3 + 2] = sz * inv - 0.5f;  // z (D axis)
}

extern "C" void kernel_launch(void* const* d_in, const int* in_sizes, int n_in,
                              void* d_out, int out_size, void* d_ws, size_t ws_size,
                              hipStream_t stream) {
    (void)in_sizes; (void)n_in; (void)out_size; (void)ws_size;
    const float* heatmaps = (const float*)d_in[0];
    float* ws  = (float*)d_ws;   // NMAPS * SPLIT * 5 floats = 60 KB
    float* out = (float*)d_out;  // NMAPS * 3 floats, fp32

    jir_partial_kernel<<<NMAPS * SPLIT, TPB, 0, stream>>>(heatmaps, ws);
    jir_final_kernel<<<(NMAPS + 63) / 64, 64, 0, stream>>>(ws, out);
}